// RAVQ4DD_51324859187283
// MI455X (gfx1250) — compile-verified
//
#include <hip/hip_runtime.h>
#include <math.h>

#define N_NODES 20000
#define N_EDGES 320000
#define IN_CH 128
#define HID 64
#define HEADS 4
#define HH 256
#define OUT_CH 40
#define N_REL 4
#define N_TYPES 3
#define N_CODES 64
#define N_RES 3
#define N_LAYERS 3

typedef float v2f __attribute__((ext_vector_type(2)));
typedef float v8f __attribute__((ext_vector_type(8)));

// Force global (address_space(1)) loads so the backend emits global_load_*
// (LOADcnt only) instead of flat_load_* (LOADcnt + DScnt false dependency).
__device__ __forceinline__ float ldg(const float* p)
{
    return *(const __attribute__((address_space(1))) float*)p;
}
__device__ __forceinline__ v2f ldg2(const float* p)
{
    return *(const __attribute__((address_space(1))) v2f*)p;
}

// ---------------------------------------------------------------------------
// fp32 GEMM on V_WMMA_F32_16X16X4_F32. One wave computes a 32x64 C macro-tile
// (2 row-tiles x 4 col-tiles = 8 accumulators); each B fragment feeds two
// WMMAs. Requires M % 32 == 0 (true for all calls: M = 20000). N handled by
// clamped loads + guarded stores. B rows walked with incrementing pointers
// (no per-iteration 64-bit multiply). Epilogue: + bias[col] + bias2[col] +
// add0[row,col]; act: 0=none, 1=exact GELU, 2=leaky(slope).
// ---------------------------------------------------------------------------
__global__ void gemm_f32_wmma(const float* __restrict__ A, int lda,
                              const float* __restrict__ B, int ldb,
                              const float* __restrict__ bias,
                              const float* __restrict__ bias2,
                              const float* __restrict__ add0, int ldadd,
                              float* __restrict__ C, int ldc,
                              int M, int N, int K, int act, float slope)
{
    const int lane   = threadIdx.x & 31;
    const int wave   = threadIdx.x >> 5;
    const int tilesN = (N + 63) >> 6;                  // 64-wide strips
    const int nTiles = (M >> 5) * tilesN;              // 32-row macro tiles
    const int tile   = blockIdx.x * (int)(blockDim.x >> 5) + wave;
    if (tile >= nTiles) return;                        // wave-uniform exit

    const int row0  = (tile / tilesN) << 5;
    const int col0  = (tile % tilesN) << 6;
    const int kHi   = (lane >> 4) << 1;                // 0 | 2 per half-wave
    const int rBase = (lane >> 4) << 3;                // C row offset (+8)

    // per-strip column indices (clamped for always-in-bounds loads)
    int cc[4], ccl[4];
#pragma unroll
    for (int j = 0; j < 4; ++j) {
        cc[j]  = col0 + (j << 4) + (lane & 15);
        ccl[j] = cc[j] < N ? cc[j] : (N - 1);
    }

    const float* Ap0 = A + (size_t)(row0 + (lane & 15)) * lda + kHi;
    const float* Ap1 = Ap0 + (size_t)16 * lda;
    v8f acc[2][4] = {{{}, {}, {}, {}}, {{}, {}, {}, {}}};

#if __has_builtin(__builtin_amdgcn_wmma_f32_16x16x4_f32)
    // incrementing B pointers: rows (k+kHi) and (k+kHi+1), 4 columns each
    const float* pB0[4];
    const float* pB1[4];
#pragma unroll
    for (int j = 0; j < 4; ++j) {
        pB0[j] = B + (size_t)kHi * ldb + ccl[j];
        pB1[j] = pB0[j] + ldb;
    }
    const size_t bStep = (size_t)4 * ldb;

    for (int k = 0; k < K; k += 4) {
        v2f a0 = ldg2(Ap0 + k);                        // global_load_b64
        v2f a1 = ldg2(Ap1 + k);                        // global_load_b64
#pragma unroll
        for (int j = 0; j < 4; ++j) {
            v2f b;
            b.x = ldg(pB0[j]);                         // global_load_b32
            b.y = ldg(pB1[j]);                         // global_load_b32
            pB0[j] += bStep;
            pB1[j] += bStep;
            acc[0][j] = __builtin_amdgcn_wmma_f32_16x16x4_f32(
                false, a0, false, b, (short)0, acc[0][j], false, false);
            acc[1][j] = __builtin_amdgcn_wmma_f32_16x16x4_f32(
                false, a1, false, b, (short)0, acc[1][j], false, false);
        }
    }
#else
    for (int i = 0; i < 2; ++i)
        for (int j = 0; j < 4; ++j)
            for (int r = 0; r < 8; ++r) {
                int rr = row0 + i * 16 + r + rBase;
                float sum = 0.f;
                for (int k = 0; k < K; ++k)
                    sum += A[(size_t)rr * lda + k] * B[(size_t)k * ldb + ccl[j]];
                acc[i][j][r] = sum;
            }
#endif

#pragma unroll
    for (int j = 0; j < 4; ++j) {
        if (cc[j] < N) {
            float bs = 0.f;
            if (bias)  bs += ldg(bias + cc[j]);
            if (bias2) bs += ldg(bias2 + cc[j]);
#pragma unroll
            for (int i = 0; i < 2; ++i) {
#pragma unroll
                for (int r = 0; r < 8; ++r) {
                    int rr = row0 + i * 16 + r + rBase;
                    float v = acc[i][j][r] + bs;
                    if (add0) v += ldg(add0 + (size_t)rr * ldadd + cc[j]);
                    if (act == 1)
                        v = 0.5f * v * (1.0f + erff(v * 0.7071067811865475f));
                    else if (act == 2)
                        v = (v >= 0.f) ? v : slope * v;
                    C[(size_t)rr * ldc + cc[j]] = v;
                }
            }
        }
    }
}

// ---------------------------------------------------------------------------
__global__ void fill_kernel(float* p, float v, long long n)
{
    long long i = (long long)blockIdx.x * blockDim.x + threadIdx.x;
    if (i < n) p[i] = v;
}

__global__ void axpy_kernel(float* __restrict__ y, const float* __restrict__ x,
                            long long n)
{
    long long i = (long long)blockIdx.x * blockDim.x + threadIdx.x;
    if (i < n) y[i] += x[i];
}

// curr[n,c] = proj[node_type[n]][n,c]
__global__ void select_type_kernel(const float* __restrict__ proj,
                                   const int* __restrict__ node_type,
                                   float* __restrict__ curr)
{
    int i = blockIdx.x * blockDim.x + threadIdx.x;
    if (i >= N_NODES * HH) return;
    int node = i / HH;
    curr[i] = proj[(size_t)node_type[node] * N_NODES * HH + i];
}

// LayerNorm over D (blockDim == D, one block per row); slope >= 0 applies
// LeakyReLU after the affine LN (reference applies leaky(LN(x))).
__global__ void ln_kernel(const float* __restrict__ in,
                          const float* __restrict__ w,
                          const float* __restrict__ b,
                          float* __restrict__ out, int D, float slope)
{
    __shared__ float s1[256];
    __shared__ float s2[256];
    const int row = blockIdx.x;
    const int t = threadIdx.x;
    float x = in[(size_t)row * D + t];
    s1[t] = x; s2[t] = x * x;
    __syncthreads();
    for (int off = D >> 1; off > 0; off >>= 1) {
        if (t < off) { s1[t] += s1[t + off]; s2[t] += s2[t + off]; }
        __syncthreads();
    }
    float mu  = s1[0] / (float)D;
    float var = s2[0] / (float)D - mu * mu;
    float y = (x - mu) * rsqrtf(var + 1e-5f) * w[t] + b[t];
    if (slope >= 0.f) y = (y >= 0.f) ? y : slope * y;
    out[(size_t)row * D + t] = y;
}

__device__ inline void atomicMaxF(float* addr, float val)
{
    if (val >= 0.f) atomicMax((int*)addr, __float_as_int(val));
    else            atomicMin((unsigned int*)addr, __float_as_uint(val));
}

// score[e,h] = 0.125 * dot(q[dst,h,:], k[src,h,:]); also atomic-max into m.
__global__ void edge_score_kernel(const int* __restrict__ ei,
                                  const int* __restrict__ et,
                                  const float* __restrict__ q,
                                  const float* __restrict__ k,
                                  float* __restrict__ es,
                                  float* __restrict__ m, int rel)
{
    long long gid = (long long)blockIdx.x * blockDim.x + threadIdx.x;
    if (gid >= (long long)N_EDGES * HEADS) return;
    int e = (int)(gid >> 2);
    int h = (int)(gid & 3);
    if (et[e] != rel) return;
    int src = ei[e];
    int dst = ei[N_EDGES + e];
    const float* qp = q + (size_t)dst * HH + h * HID;
    const float* kp = k + (size_t)src * HH + h * HID;
    float acc = 0.f;
    for (int c = 0; c < HID; ++c) acc += ldg(qp + c) * ldg(kp + c);
    acc *= 0.125f;                               // 1/sqrt(64)
    es[(size_t)e * HEADS + h] = acc;
    atomicMaxF(&m[dst * HEADS + h], acc);
}

// es[e,h] = exp(score - m[seg]); atomic-add into s[seg].
__global__ void edge_exp_kernel(const int* __restrict__ ei,
                                const int* __restrict__ et,
                                float* __restrict__ es,
                                const float* __restrict__ m,
                                float* __restrict__ s, int rel)
{
    long long gid = (long long)blockIdx.x * blockDim.x + threadIdx.x;
    if (gid >= (long long)N_EDGES * HEADS) return;
    int e = (int)(gid >> 2);
    int h = (int)(gid & 3);
    if (et[e] != rel) return;
    int dst = ei[N_EDGES + e];
    float ev = expf(es[(size_t)e * HEADS + h] - m[dst * HEADS + h]);
    es[(size_t)e * HEADS + h] = ev;
    atomicAdd(&s[dst * HEADS + h], ev);
}

// agg[dst, :] += v[src, :] * alpha  (one block of 256 threads per edge)
__global__ void edge_agg_kernel(const int* __restrict__ ei,
                                const int* __restrict__ et,
                                const float* __restrict__ v,
                                const float* __restrict__ es,
                                const float* __restrict__ s,
                                float* __restrict__ agg, int rel)
{
    int e = blockIdx.x;
    if (et[e] != rel) return;
    int t = threadIdx.x;
    int h = t >> 6;
    int src = ei[e];
    int dst = ei[N_EDGES + e];
    float alpha = es[(size_t)e * HEADS + h] / (s[dst * HEADS + h] + 1e-16f);
    atomicAdd(&agg[(size_t)dst * HH + t], ldg(v + (size_t)src * HH + t) * alpha);
}

// Residual VQ: one 64-thread block per node. Writes quant into idc columns
// [layer*64, layer*64+64), indices (as float) into out_idx[n*9 + layer*3 + l],
// and atomically accumulates 1.25*mean(min_dist) into *loss.
__global__ void vq_kernel(const float* __restrict__ hc,
                          const float* __restrict__ cb,
                          float* __restrict__ idc,
                          float* __restrict__ out_idx,
                          float* __restrict__ loss, int layer)
{
    __shared__ float r[HID];
    __shared__ float qa[HID];
    __shared__ float dist[N_CODES];
    __shared__ int   sidx[N_CODES];
    const int n = blockIdx.x;
    const int t = threadIdx.x;                  // 0..63
    r[t]  = hc[(size_t)n * HID + t];
    qa[t] = 0.f;
    __syncthreads();
    float lsum = 0.f;
    for (int l = 0; l < N_RES; ++l) {
        const float* cbl = cb + (size_t)l * N_CODES * HID;
        float d = 0.f;
        for (int c = 0; c < HID; ++c) {
            float df = r[c] - ldg(cbl + (size_t)t * HID + c);
            d += df * df;
        }
        dist[t] = d; sidx[t] = t;
        __syncthreads();
        for (int off = 32; off > 0; off >>= 1) {
            if (t < off) {
                if (dist[t + off] < dist[t] ||
                    (dist[t + off] == dist[t] && sidx[t + off] < sidx[t])) {
                    dist[t] = dist[t + off]; sidx[t] = sidx[t + off];
                }
            }
            __syncthreads();
        }
        int best = sidx[0];
        float dmin = dist[0];
        __syncthreads();
        float qc = ldg(cbl + (size_t)best * HID + t);
        qa[t] += qc;
        r[t]  -= qc;
        __syncthreads();
        if (t == 0) {
            out_idx[(size_t)n * (N_LAYERS * N_RES) + layer * N_RES + l] = (float)best;
            lsum += dmin;
        }
    }
    if (t == 0)
        atomicAdd(loss, 1.25f * lsum / (float)((long long)N_NODES * HID));
    idc[(size_t)n * (N_LAYERS * HID) + layer * HID + t] = qa[t];
}

// ---------------------------------------------------------------------------
extern "C" void kernel_launch(void* const* d_in, const int* in_sizes, int n_in,
                              void* d_out, int out_size, void* d_ws, size_t ws_size,
                              hipStream_t stream)
{
    (void)in_sizes; (void)n_in; (void)out_size; (void)ws_size;

    // ---- inputs (setup_inputs insertion order, params flattened in dict order)
    const float* x  = (const float*)d_in[0];
    const int*   ei = (const int*)d_in[1];    // [2,E]: src then dst
    const int*   et = (const int*)d_in[2];
    const int*   nt = (const int*)d_in[3];
    const float* tp_W = (const float*)d_in[4];
    const float* tp_b = (const float*)d_in[5];
    struct LP { const float *ln_w,*ln_b,*Wq,*Wk,*Wv,*root_W,*root_b,*bias,
                            *WO_W,*WO_b,*res_W,*res_b,*cb; };
    LP L[N_LAYERS];
    for (int l = 0; l < N_LAYERS; ++l) {
        int b = 6 + l * 13;
        L[l].ln_w  = (const float*)d_in[b + 0];
        L[l].ln_b  = (const float*)d_in[b + 1];
        L[l].Wq    = (const float*)d_in[b + 2];
        L[l].Wk    = (const float*)d_in[b + 3];
        L[l].Wv    = (const float*)d_in[b + 4];
        L[l].root_W= (const float*)d_in[b + 5];
        L[l].root_b= (const float*)d_in[b + 6];
        L[l].bias  = (const float*)d_in[b + 7];
        L[l].WO_W  = (const float*)d_in[b + 8];
        L[l].WO_b  = (const float*)d_in[b + 9];
        L[l].res_W = (const float*)d_in[b + 10];
        L[l].res_b = (const float*)d_in[b + 11];
        L[l].cb    = (const float*)d_in[b + 12];
    }
    int cb0 = 6 + N_LAYERS * 13;   // 45
    const float* c_W1   = (const float*)d_in[cb0 + 0];
    const float* c_b1   = (const float*)d_in[cb0 + 1];
    const float* rb_ln1w= (const float*)d_in[cb0 + 2];
    const float* rb_ln1b= (const float*)d_in[cb0 + 3];
    const float* rb_W1  = (const float*)d_in[cb0 + 4];
    const float* rb_b1  = (const float*)d_in[cb0 + 5];
    const float* rb_ln2w= (const float*)d_in[cb0 + 6];
    const float* rb_ln2b= (const float*)d_in[cb0 + 7];
    const float* rb_W2  = (const float*)d_in[cb0 + 8];
    const float* rb_b2  = (const float*)d_in[cb0 + 9];
    const float* c_lnw  = (const float*)d_in[cb0 + 10];
    const float* c_lnb  = (const float*)d_in[cb0 + 11];
    const float* c_W2   = (const float*)d_in[cb0 + 12];
    const float* c_b2   = (const float*)d_in[cb0 + 13];
    const float* aux_W1 = (const float*)d_in[cb0 + 14];
    const float* aux_b1 = (const float*)d_in[cb0 + 15];
    const float* aux_W2 = (const float*)d_in[cb0 + 16];
    const float* aux_b2 = (const float*)d_in[cb0 + 17];
    const float* pred_W = (const float*)d_in[cb0 + 18];
    const float* pred_b = (const float*)d_in[cb0 + 19];

    // ---- workspace layout (floats)
    float* W = (float*)d_ws;
    const size_t SZ = (size_t)N_NODES * HH;        // 5,120,000
    float* curr = W;
    float* xn   = W + 1 * SZ;
    float* q    = W + 2 * SZ;                      // also proj[0]
    float* kbuf = W + 3 * SZ;                      // also proj[1]
    float* vbuf = W + 4 * SZ;                      // also proj[2]
    float* agg  = W + 5 * SZ;
    float* t1   = W + 6 * SZ;
    float* hc   = W + 7 * SZ;
    float* xloc = hc   + (size_t)N_NODES * HID;
    float* idc  = xloc + (size_t)N_NODES * HID;
    float* es   = idc  + (size_t)N_NODES * (N_LAYERS * HID);
    float* mbuf = es   + (size_t)N_EDGES * HEADS;
    float* sbuf = mbuf + (size_t)N_NODES * HEADS;
    float* h1   = sbuf + (size_t)N_NODES * HEADS;
    float* t2   = h1   + (size_t)N_NODES * 128;
    float* t3   = t2   + (size_t)N_NODES * 128;
    float* a1   = t3   + (size_t)N_NODES * 128;

    // ---- output layout (floats): id_out, aux, gnn, loss, idxs
    float* out_id   = (float*)d_out;
    float* out_aux  = out_id  + (size_t)N_NODES * OUT_CH;
    float* out_gnn  = out_aux + (size_t)N_NODES * OUT_CH;
    float* out_loss = out_gnn + (size_t)N_NODES * OUT_CH;
    float* out_idx  = out_loss + 1;

    auto fill = [&](float* p, float val, size_t n) {
        fill_kernel<<<(unsigned)((n + 255) / 256), 256, 0, stream>>>(p, val, (long long)n);
    };
    auto gemm = [&](const float* A, int lda, const float* B, int ldb,
                    const float* bias, const float* bias2,
                    const float* add0, int ldadd,
                    float* C, int ldc, int M, int N, int K, int act, float slope) {
        int tiles = (M / 32) * ((N + 63) / 64);
        int blocks = (tiles + 7) / 8;              // 8 waves (256 thr) per block
        gemm_f32_wmma<<<blocks, 256, 0, stream>>>(A, lda, B, ldb, bias, bias2,
                                                  add0, ldadd, C, ldc, M, N, K,
                                                  act, slope);
    };

    // ================= TypeSpecificProjector =================
    for (int t = 0; t < N_TYPES; ++t)
        gemm(x, IN_CH, tp_W + (size_t)t * IN_CH * HH, HH,
             tp_b + (size_t)t * HH, nullptr, nullptr, 0,
             q + (size_t)t * SZ, HH, N_NODES, HH, IN_CH, 0, 0.f);
    select_type_kernel<<<(N_NODES * HH + 255) / 256, 256, 0, stream>>>(q, nt, curr);

    fill(xloc, 0.f, (size_t)N_NODES * HID);
    fill(out_loss, 0.f, 1);

    // ================= 3 RelationAware layers =================
    for (int l = 0; l < N_LAYERS; ++l) {
        const LP& p = L[l];
        ln_kernel<<<N_NODES, HH, 0, stream>>>(curr, p.ln_w, p.ln_b, xn, HH, -1.f);
        fill(agg, 0.f, SZ);
        for (int rel = 0; rel < N_REL; ++rel) {
            const size_t wo = (size_t)rel * HH * HH;
            gemm(xn, HH, p.Wq + wo, HH, nullptr, nullptr, nullptr, 0,
                 q,    HH, N_NODES, HH, HH, 0, 0.f);
            gemm(xn, HH, p.Wk + wo, HH, nullptr, nullptr, nullptr, 0,
                 kbuf, HH, N_NODES, HH, HH, 0, 0.f);
            gemm(xn, HH, p.Wv + wo, HH, nullptr, nullptr, nullptr, 0,
                 vbuf, HH, N_NODES, HH, HH, 0, 0.f);
            fill(mbuf, -INFINITY, (size_t)N_NODES * HEADS);
            fill(sbuf, 0.f,       (size_t)N_NODES * HEADS);
            const int eg = (N_EDGES * HEADS + 255) / 256;
            edge_score_kernel<<<eg, 256, 0, stream>>>(ei, et, q, kbuf, es, mbuf, rel);
            edge_exp_kernel  <<<eg, 256, 0, stream>>>(ei, et, es, mbuf, sbuf, rel);
            edge_agg_kernel  <<<N_EDGES, 256, 0, stream>>>(ei, et, vbuf, es, sbuf, agg, rel);
        }
        // total = xn@root_W + root_b + bias + agg
        gemm(xn, HH, p.root_W, HH, p.root_b, p.bias, agg, HH,
             t1, HH, N_NODES, HH, HH, 0, 0.f);
        // hc = gelu(total @ WO_W + WO_b)
        gemm(t1, HH, p.WO_W, HID, p.WO_b, nullptr, nullptr, 0,
             hc, HID, N_NODES, HID, HH, 1, 0.f);
        axpy_kernel<<<(N_NODES * HID + 255) / 256, 256, 0, stream>>>(
            xloc, hc, (long long)N_NODES * HID);
        vq_kernel<<<N_NODES, N_CODES, 0, stream>>>(hc, p.cb, idc, out_idx, out_loss, l);
        // curr = curr + hc @ res_W + res_b
        gemm(hc, HID, p.res_W, HH, p.res_b, nullptr, curr, HH,
             curr, HH, N_NODES, HH, HID, 0, 0.f);
    }

    // ================= id classifier =================
    gemm(idc, N_LAYERS * HID, c_W1, 128, c_b1, nullptr, nullptr, 0,
         h1, 128, N_NODES, 128, N_LAYERS * HID, 0, 0.f);
    ln_kernel<<<N_NODES, 128, 0, stream>>>(h1, rb_ln1w, rb_ln1b, t3, 128, 0.1f);
    gemm(t3, 128, rb_W1, 128, rb_b1, nullptr, nullptr, 0,
         t2, 128, N_NODES, 128, 128, 0, 0.f);
    ln_kernel<<<N_NODES, 128, 0, stream>>>(t2, rb_ln2w, rb_ln2b, t3, 128, 0.1f);
    gemm(t3, 128, rb_W2, 128, rb_b2, nullptr, h1, 128,
         h1, 128, N_NODES, 128, 128, 0, 0.f);          // h = r + t
    ln_kernel<<<N_NODES, 128, 0, stream>>>(h1, c_lnw, c_lnb, t3, 128, 0.1f);
    gemm(t3, 128, c_W2, OUT_CH, c_b2, nullptr, nullptr, 0,
         out_id, OUT_CH, N_NODES, OUT_CH, 128, 0, 0.f);

    // ================= aux head (uses last layer's hc) =================
    gemm(hc, HID, aux_W1, HID, aux_b1, nullptr, nullptr, 0,
         a1, HID, N_NODES, HID, HID, 2, 0.01f);
    gemm(a1, HID, aux_W2, OUT_CH, aux_b2, nullptr, nullptr, 0,
         out_aux, OUT_CH, N_NODES, OUT_CH, HID, 0, 0.f);

    // ================= gnn head =================
    gemm(xloc, HID, pred_W, OUT_CH, pred_b, nullptr, nullptr, 0,
         out_gnn, OUT_CH, N_NODES, OUT_CH, HID, 0, 0.f);
}